// QuantumWrapper_65481071400865
// MI455X (gfx1250) — compile-verified
//
#include <hip/hip_runtime.h>

typedef float v4f __attribute__((ext_vector_type(4)));

// out[i] = sin(a)*sin(b) for interleaved (a,b) float32 pairs.
// Memory-bound streaming kernel: 12 B/elem, ~201 MB total, ~8.6 us at 23.3 TB/s.
// Each thread: 2x global_load_b128 (8 input floats = 4 pairs), 8x v_sin_f32,
// 4x v_mul_f32, 1x global_store_b128. Non-temporal hints keep the zero-reuse
// stream from thrashing the 192 MB L2.

__device__ __forceinline__ float hw_sin(float x) {
    // v_sin_f32 computes sin(2*pi*s); pre-scale by 1/(2*pi).
    return __builtin_amdgcn_sinf(x * 0.15915494309189535f);
}

__global__ __launch_bounds__(256) void sinprod_vec4_kernel(
    const v4f* __restrict__ in,   // viewed as float4: two (a,b) pairs per float4
    v4f* __restrict__ out,        // 4 outputs per float4
    int n4)                       // number of float4 outputs
{
    int i = blockIdx.x * blockDim.x + threadIdx.x;
    if (i >= n4) return;

    v4f p0 = __builtin_nontemporal_load(&in[2 * i + 0]); // pairs 4i, 4i+1
    v4f p1 = __builtin_nontemporal_load(&in[2 * i + 1]); // pairs 4i+2, 4i+3

    v4f r;
    r.x = hw_sin(p0.x) * hw_sin(p0.y);
    r.y = hw_sin(p0.z) * hw_sin(p0.w);
    r.z = hw_sin(p1.x) * hw_sin(p1.y);
    r.w = hw_sin(p1.z) * hw_sin(p1.w);

    __builtin_nontemporal_store(r, &out[i]);
}

// Scalar tail kernel (only launched if out_size % 4 != 0; unused for N=2^24
// but kept so kernel_launch is correct for any size).
__global__ __launch_bounds__(64) void sinprod_tail_kernel(
    const float* __restrict__ in, float* __restrict__ out, int start, int n)
{
    int i = start + blockIdx.x * blockDim.x + threadIdx.x;
    if (i >= n) return;
    out[i] = hw_sin(in[2 * i]) * hw_sin(in[2 * i + 1]);
}

extern "C" void kernel_launch(void* const* d_in, const int* in_sizes, int n_in,
                              void* d_out, int out_size, void* d_ws, size_t ws_size,
                              hipStream_t stream) {
    (void)in_sizes; (void)n_in; (void)d_ws; (void)ws_size;
    const float* in = (const float*)d_in[0];
    float* out = (float*)d_out;

    int n4 = out_size >> 2;          // 4,194,304 for the reference shape
    int rem = out_size & 3;          // 0 for the reference shape

    if (n4 > 0) {
        const int block = 256;
        const int grid = (n4 + block - 1) / block;   // 16,384 blocks
        sinprod_vec4_kernel<<<grid, block, 0, stream>>>(
            (const v4f*)in, (v4f*)out, n4);
    }
    if (rem > 0) {
        sinprod_tail_kernel<<<1, 64, 0, stream>>>(in, out, n4 * 4, out_size);
    }
}